// End2EndDP_22471268893091
// MI455X (gfx1250) — compile-verified
//
#include <hip/hip_runtime.h>
#include <math.h>

// CDNA5 wave32 WMMA fragments for V_WMMA_F32_16X16X4_F32
typedef __attribute__((ext_vector_type(2))) float v2f;   // A (16x4 f32) / B (4x16 f32): 64 vals / 32 lanes
typedef __attribute__((ext_vector_type(8))) float v8f;   // C/D (16x16 f32): 256 vals / 32 lanes

#define LSEQ     512
#define PADL     544      // 514 rounded up; padding stays BIG
#define NTHREADS 512      // 16 wave32s per workgroup
#define NWAVES   16
#define BIGF     1e30f

// Soft-DTW banded DP, one workgroup per batch chain.
// Per row i:  softmin[j] = m - log( sum_p K[j,p] * exp(-(prev[p]-m)) ),
// K[j,p] = 1 (p<j) or e^{-(p-j+1)} (p>=j)  -> banded fp32 WMMA GEMM.
__global__ __launch_bounds__(NTHREADS)
void End2EndDP_22471268893091_kernel(const float* __restrict__ Dmat,
                                     const int* __restrict__ len_a,
                                     const int* __restrict__ len_b,
                                     float* __restrict__ out)
{
    __shared__ float prev[PADL];   // DP row i-1
    __shared__ float xv[PADL];     // exp(-(prev - m))
    __shared__ float sv[PADL];     // S = K @ xv
    __shared__ float ktab[128];    // ktab[t]: t=0 -> 1 (p<j); t=1..96 -> e^{-t}; t>96 -> 0
    __shared__ float red[NWAVES];

    const int b     = blockIdx.x;
    const int tid   = threadIdx.x;
    const int lane  = tid & 31;
    const int wave  = tid >> 5;
    const int jlane = lane & 15;                     // N / M-row position inside fragment
    const int ksel  = (lane >> 4) << 1;              // A: lanes<16 -> K{0,1}, lanes>=16 -> K{2,3}
    const float n0f = (jlane == 0) ? 1.0f : 0.0f;    // only B column N=0 carries x

    const int lenA = len_a[b];
    const int lenB = len_b[b];
    if (lenA < 1) { if (tid == 0) out[b] = 0.0f; return; }

    int ad = lenA - lenB; if (ad < 0) ad = -ad;
    int ws = ad + 1; if (ws < 5) ws = 5;

    const float Mf = (float)lenA, Nf = (float)lenB;
    const float dscale = 5.0f / Mf;                  // LBD_DUR / M
    const float lenpen = fabsf(Mf - Nf) / Mf * 0.2f; // |M-N|/M * LBD_LEN

    for (int p = tid; p < PADL; p += NTHREADS) prev[p] = (p == 0) ? 0.0f : BIGF;
    if (tid < 128)
        ktab[tid] = (tid == 0) ? 1.0f : ((tid <= 96) ? expf(-(float)tid) : 0.0f);
    __syncthreads();

    int kLo = 0, kHi = 1;            // range of possibly-finite prev entries

    for (int i = 1; i <= lenA; ++i) {
        int lb = i - ws; if (lb < 1) lb = 1;
        int rb = i + ws; if (rb > lenB + 2) rb = lenB + 2;

        // ---- phase 1: m = min(prev) over live band ----
        float local = BIGF;
        for (int p = kLo + tid; p < kHi; p += NTHREADS) local = fminf(local, prev[p]);
        #pragma unroll
        for (int off = 16; off > 0; off >>= 1)
            local = fminf(local, __shfl_xor(local, off, 32));
        if (lane == 0) red[wave] = local;
        __syncthreads();
        if (tid == 0) {
            float mm = red[0];
            #pragma unroll
            for (int w = 1; w < NWAVES; ++w) mm = fminf(mm, red[w]);
            red[0] = mm;
        }
        __syncthreads();
        const float m = red[0];

        // ---- phase 2: x[p] = exp(-(prev[p]-m)) over the k-tile span (BIG -> exact 0) ----
        const int kt0  = kLo >> 2;
        const int kt1g = (kHi - 1) >> 2;
        const int pEnd = kt1g * 4 + 4;               // <= 516 < PADL
        for (int p = kt0 * 4 + tid; p < pEnd; p += NTHREADS)
            xv[p] = expf(-(prev[p] - m));
        __syncthreads();

        // ---- phase 3: banded GEMM S = K @ x via fp32 WMMA 16x16x4 (branchless frags) ----
        const int jt0 = lb >> 4, jt1 = (rb - 1) >> 4;
        for (int jt = jt0 + wave; jt <= jt1; jt += NWAVES) {
            v8f c = {};
            const int j = jt * 16 + jlane;
            // tiles with p - j >= 96 for the whole tile contribute exact zero: cap kt
            const int ktcap = (jt * 16 + 110) >> 2;  // uniform per wave
            const int kt1   = (kt1g < ktcap) ? kt1g : ktcap;
            int dbase = kt0 * 4 + ksel - j + 1;      // = d0 + 1 at kt0
            for (int kt = kt0; kt <= kt1; ++kt, dbase += 4) {
                const int p0 = kt * 4 + ksel;
                int t0 = dbase;     t0 = (t0 < 0) ? 0 : t0; t0 = (t0 > 127) ? 127 : t0;
                int t1 = dbase + 1; t1 = (t1 < 0) ? 0 : t1; t1 = (t1 > 127) ? 127 : t1;
                v2f a;  a.x  = ktab[t0];        a.y  = ktab[t1];
                v2f bb; bb.x = xv[p0] * n0f;    bb.y = xv[p0 + 1] * n0f;
                c = __builtin_amdgcn_wmma_f32_16x16x4_f32(
                        false, a, false, bb, (short)0, c, false, false);
            }
            // column N=0 of D-tile lives in lane 0 (M=g) and lane 16 (M=g+8)
            if (jlane == 0) {
                const int mb = jt * 16 + ((lane >> 4) ? 8 : 0);
                #pragma unroll
                for (int g = 0; g < 8; ++g) sv[mb + g] = c[g];
            }
        }
        __syncthreads();

        // ---- phase 4: new row = D + length_dist + softmin, banded mask ----
        const float fi = (float)i;
        for (int j = tid; j < PADL; j += NTHREADS) {
            float nv = BIGF;
            if (j >= lb && j < rb) {
                const float s = sv[j];
                if (s > 0.0f) {
                    const float softmin = m - logf(s);
                    const float fj   = (float)j;
                    const float diff = fi - fj;
                    const float base = 1.0f - expf(-(diff * diff) / (2.0f * (fj + 1.0f)));
                    const float dv   = Dmat[((size_t)b * LSEQ + (size_t)(i - 1)) * LSEQ + (size_t)(j - 1)];
                    nv = dv + base * dscale + lenpen + softmin;
                    if (!(nv < BIGF)) nv = BIGF;     // clamp inf/nan to BIG sentinel
                }
            }
            if (i == lenA && j == lenB) out[b] = nv; // ans = window[b, len_a, len_b]
            prev[j] = nv;
        }
        __syncthreads();

        kLo = lb; kHi = rb;
    }
}

extern "C" void kernel_launch(void* const* d_in, const int* in_sizes, int n_in,
                              void* d_out, int out_size, void* d_ws, size_t ws_size,
                              hipStream_t stream) {
    (void)n_in; (void)d_ws; (void)ws_size; (void)out_size;
    const float* Dmat  = (const float*)d_in[0];   // [B, 512, 512] fp32
    const int*   len_a = (const int*)d_in[1];     // [B] int32
    const int*   len_b = (const int*)d_in[2];     // [B] int32
    float*       out   = (float*)d_out;           // [B] fp32
    const int B = in_sizes[1];
    End2EndDP_22471268893091_kernel<<<dim3(B), dim3(NTHREADS), 0, stream>>>(
        Dmat, len_a, len_b, out);
}